// RPN_15607911153787
// MI455X (gfx1250) — compile-verified
//
#include <hip/hip_runtime.h>
#include <math.h>

// Batched greedy NMS for MI455X (gfx1250).
// One workgroup per image; boxes+scores staged into LDS via CDNA5 async
// global->LDS copies (ASYNCcnt path), bitonic sort + 30 greedy suppression
// sweeps entirely in LDS.

#define B_IMGS   16
#define N_PROP   2000
#define N_PAD    2048
#define N_KEEP   30
#define IOU_TH   0.3f
#define EPS_N    0.05f
#define NTHREADS 1024

__global__ __launch_bounds__(NTHREADS)
void rpn_nms_kernel(const float* __restrict__ proposals,   // [B, N, 4] xyxy
                    const float* __restrict__ scores,      // [B, N]
                    float* __restrict__ out)               // [B, 30, 5]
{
    __shared__ float4 s_box[N_PAD];     // boxes in ORIGINAL order
    __shared__ float  s_sc[N_PAD];      // scores, sorted in-place by bitonic sort
    __shared__ int    s_id[N_PAD];      // rank -> original index after sort
    __shared__ int    s_sup[N_PAD];     // suppressed flags (by rank)
    __shared__ int    s_picked[N_KEEP];
    __shared__ float  s_kept[N_KEEP];
    __shared__ int    s_pick;
    __shared__ float  s_mn, s_mx;

    const int b   = blockIdx.x;
    const int tid = threadIdx.x;

    const float4* gbox = (const float4*)(proposals + (size_t)b * N_PROP * 4);
    const float*  gsc  = scores + (size_t)b * N_PROP;

    // ---- Phase 0: async bulk copy global -> LDS (CDNA5 ASYNCcnt path) ----
    for (int r = tid; r < N_PROP; r += NTHREADS) {
        unsigned lbox = (unsigned)(unsigned long long)&s_box[r];   // low 32 bits == LDS offset
        unsigned long long gb = (unsigned long long)(gbox + r);
        asm volatile("global_load_async_to_lds_b128 %0, %1, off"
                     :: "v"(lbox), "v"(gb) : "memory");
        unsigned lsc = (unsigned)(unsigned long long)&s_sc[r];
        unsigned long long gs = (unsigned long long)(gsc + r);
        asm volatile("global_load_async_to_lds_b32 %0, %1, off"
                     :: "v"(lsc), "v"(gs) : "memory");
    }
    // init ids / suppression / pad entries (disjoint from async destinations or pad-only)
    for (int r = tid; r < N_PAD; r += NTHREADS) {
        s_id[r]  = r;
        s_sup[r] = 0;
        if (r >= N_PROP) {
            s_sc[r]  = -INFINITY;                 // pads sort to ranks >= N_PROP
            s_box[r] = make_float4(0.f, 0.f, 0.f, 0.f);
        }
    }
    asm volatile("s_wait_asynccnt 0" ::: "memory"); // drain this wave's async copies
    __syncthreads();                                 // make them visible WG-wide

    // ---- Phase 1: bitonic sort of 2048 (score,idx) pairs ----
    // Order: score descending, ties by original index ascending
    // (== stable jnp.argsort(-scores)).
    for (unsigned k = 2; k <= N_PAD; k <<= 1) {
        for (unsigned j = k >> 1; j > 0; j >>= 1) {
            unsigned t = (unsigned)tid;
            unsigned i = 2u * j * (t / j) + (t % j);
            unsigned p = i + j;
            bool up = ((i & k) == 0u);
            float sa = s_sc[i], sb = s_sc[p];
            int   ia = s_id[i], ib = s_id[p];
            bool a_first = (sa > sb) || ((sa == sb) && (ia < ib));
            bool do_swap = up ? (!a_first) : a_first;
            if (do_swap) {
                s_sc[i] = sb; s_sc[p] = sa;
                s_id[i] = ib; s_id[p] = ia;
            }
            __syncthreads();
        }
    }

    // ---- Phase 2: greedy NMS, 30 sequential picks ----
    for (int it = 0; it < N_KEEP; ++it) {
        if (tid == 0) s_pick = 0x7FFFFFFF;
        __syncthreads();

        // highest-scoring survivor == lowest unsuppressed rank (scores sorted desc)
        int local = 0x7FFFFFFF;
        for (int r = tid; r < N_PROP; r += NTHREADS)
            if (!s_sup[r] && r < local) local = r;
        atomicMin(&s_pick, local);
        __syncthreads();

        int pick = s_pick;
        if (pick == 0x7FFFFFFF) pick = 0;   // jnp.argmax over all -inf -> index 0
        if (tid == 0) s_picked[it] = pick;

        float4 pb = s_box[s_id[pick]];
        float p_area;
        {
#pragma clang fp contract(off)
            p_area = (pb.z - pb.x) * (pb.w - pb.y);
        }
        for (int r = tid; r < N_PROP; r += NTHREADS) {
            float4 cb = s_box[s_id[r]];
            int sup;
            {
#pragma clang fp contract(off)
                float carea = (cb.z - cb.x) * (cb.w - cb.y);
                float lx = fmaxf(pb.x, cb.x);
                float ly = fmaxf(pb.y, cb.y);
                float rx = fminf(pb.z, cb.z);
                float ry = fminf(pb.w, cb.w);
                float w  = rx - lx; if (w < 0.f) w = 0.f;
                float h  = ry - ly; if (h < 0.f) h = 0.f;
                float inter = w * h;
                float iou   = inter / (p_area + carea - inter);
                sup = (iou > IOU_TH) ? 1 : 0;   // pick suppresses itself (IoU==1)
            }
            if (sup) s_sup[r] = 1;
        }
        __syncthreads();
    }

    // ---- Phase 3: gather, normalize, write [30 x 5] ----
    if (tid < N_KEEP) {
        s_kept[tid] = s_sc[s_picked[tid]];
    }
    __syncthreads();
    if (tid == 0) {
        float mn = s_kept[0], mx = s_kept[0];
        for (int i = 1; i < N_KEEP; ++i) {
            mn = fminf(mn, s_kept[i]);
            mx = fmaxf(mx, s_kept[i]);
        }
        s_mn = mn; s_mx = mx;
    }
    __syncthreads();
    if (tid < N_KEEP) {
        int rank = s_picked[tid];
        float4 bx = s_box[s_id[rank]];
        float nrm = (s_kept[tid] - s_mn) / fabsf(s_mx - s_mn) + EPS_N;
        float* op = out + ((size_t)b * N_KEEP + tid) * 5;
        op[0] = bx.x; op[1] = bx.y; op[2] = bx.z; op[3] = bx.w; op[4] = nrm;
    }
}

extern "C" void kernel_launch(void* const* d_in, const int* in_sizes, int n_in,
                              void* d_out, int out_size, void* d_ws, size_t ws_size,
                              hipStream_t stream) {
    (void)in_sizes; (void)n_in; (void)d_ws; (void)ws_size; (void)out_size;
    const float* proposals = (const float*)d_in[0];   // [16, 2000, 4] f32
    const float* scores    = (const float*)d_in[1];   // [16, 2000]    f32
    float* out             = (float*)d_out;           // [16, 30, 5]   f32
    rpn_nms_kernel<<<B_IMGS, NTHREADS, 0, stream>>>(proposals, scores, out);
}